// GNN_11544872092280
// MI455X (gfx1250) — compile-verified
//
#include <hip/hip_runtime.h>
#include <hip/hip_bf16.h>
#include <math.h>

// ---------------- problem constants (match reference) ----------------
#define N_NODES  50000
#define N_EDGES  800000
#define N_GRAPHS 128
#define F_IN     128
#define HID      64
#define EDGE_DIM 16
#define H1       4
#define H2       2
#define E_TOT    (N_EDGES + N_NODES)   // edges + self loops
#define NEG_SLOPE 0.2f

typedef __attribute__((ext_vector_type(2))) float v2f;
typedef __attribute__((ext_vector_type(8))) float v8f;

// ---- order-preserving float<->uint encoding for integer atomicMax ----
__device__ __forceinline__ unsigned f2ord(float f) {
  unsigned u = __float_as_uint(f);
  return (u & 0x80000000u) ? ~u : (u | 0x80000000u);
}
__device__ __forceinline__ float ord2f(unsigned k) {
  if (k == 0u) return -3.4e38f;  // "never updated" sentinel (segments are non-empty: self loops)
  return (k & 0x80000000u) ? __uint_as_float(k & 0x7fffffffu) : __uint_as_float(~k);
}
__device__ __forceinline__ float lrelu(float x) { return x > 0.f ? x : NEG_SLOPE * x; }

// virtual concatenated edge list: e < N_EDGES -> real edge, else self loop (e - N_EDGES)
__device__ __forceinline__ int esrc(const int* ei, int e) { return e < N_EDGES ? ei[e] : (e - N_EDGES); }
__device__ __forceinline__ int edst(const int* ei, int e) { return e < N_EDGES ? ei[N_EDGES + e] : (e - N_EDGES); }
__device__ __forceinline__ float eattr(const float* ea, const float* loop, int e, int d) {
  return e < N_EDGES ? ea[(size_t)e * EDGE_DIM + d]
                     : loop[(size_t)(e - N_EDGES) * EDGE_DIM + d];
}

// ---------------- kernels ----------------

__global__ void k_zero(unsigned* p, long long n) {
  long long i = (long long)blockIdx.x * blockDim.x + threadIdx.x;
  long long st = (long long)gridDim.x * blockDim.x;
  for (; i < n; i += st) p[i] = 0u;
}

// per-dst mean of edge_attr for self-loop fill (PyG fill_value='mean')
__global__ void k_loop_accum(const float* __restrict__ ea, const int* __restrict__ ei,
                             float* cnt, float* loop) {
  long long t = (long long)blockIdx.x * blockDim.x + threadIdx.x;
  if (t >= (long long)N_EDGES * EDGE_DIM) return;
  int e = (int)(t / EDGE_DIM), d = (int)(t % EDGE_DIM);
  int dd = ei[N_EDGES + e];
  atomicAdd(&loop[(size_t)dd * EDGE_DIM + d], ea[t]);
  if (d == 0) atomicAdd(&cnt[dd], 1.0f);
}
__global__ void k_loop_div(float* loop, const float* __restrict__ cnt) {
  int t = blockIdx.x * blockDim.x + threadIdx.x;
  if (t >= N_NODES * EDGE_DIM) return;
  loop[t] /= fmaxf(cnt[t / EDGE_DIM], 1.0f);
}

// fp32 WMMA GEMM: C[M,N] = A[M,K] @ B[K,N]. One wave -> one 16x16 tile.
// A 16x4 layout: lanes 0-15 hold K={k,k+1}, lanes 16-31 K={k+2,k+3} (ISA 7.12.2).
// B 4x16: row striped across lanes, same K split across lane halves.
// C: VGPR j -> (M=j + 8*laneHalf, N=lane&15).
__global__ void k_wmma_gemm_f32(const float* __restrict__ A, const float* __restrict__ B,
                                float* __restrict__ C, int M, int N, int K) {
  int wave = (int)(((long long)blockIdx.x * blockDim.x + threadIdx.x) >> 5);
  int lane = threadIdx.x & 31;
  int nt = N >> 4;
  int tM = wave / nt, tN = wave - tM * nt;
  if (tM >= (M >> 4)) return;                    // wave-uniform guard: EXEC stays all-1s
  int laneM = lane & 15, half = lane >> 4;
  const float* Ar = A + (size_t)(tM * 16 + laneM) * K;
  int col = tN * 16 + laneM;
  v8f acc = {};
  for (int k0 = 0; k0 < K; k0 += 4) {
    int ka = k0 + half * 2;
    v2f a, b;
    a[0] = Ar[ka];
    a[1] = Ar[ka + 1];
    b[0] = B[(size_t)ka * N + col];
    b[1] = B[(size_t)(ka + 1) * N + col];
    acc = __builtin_amdgcn_wmma_f32_16x16x4_f32(false, a, false, b, (short)0, acc,
                                                false, false);
  }
#pragma unroll
  for (int j = 0; j < 8; ++j) {
    int row = tM * 16 + j + half * 8;
    C[(size_t)row * N + col] = acc[j];
  }
}

// V[d,h] = sum_c We[d, h*HID+c] * a_e[h,c]   (folds ea@We -> s_e into a tiny [16,H] matrix)
__global__ void k_fold(const float* __restrict__ We, const float* __restrict__ a_e,
                       float* V, int H) {
  int t = blockIdx.x * blockDim.x + threadIdx.x;
  if (t >= EDGE_DIM * H) return;
  int d = t / H, h = t - d * H;
  float s = 0.f;
  for (int c = 0; c < HID; ++c) s += We[(size_t)d * (H * HID) + h * HID + c] * a_e[h * HID + c];
  V[d * H + h] = s;
}

// s_src[n,h], s_dst[n,h] from xs[n,h,:]
__global__ void k_sdots(const float* __restrict__ xs, const float* __restrict__ a_src,
                        const float* __restrict__ a_dst, float* s_src, float* s_dst, int H) {
  int t = blockIdx.x * blockDim.x + threadIdx.x;
  if (t >= N_NODES * H) return;
  int n = t / H, h = t - n * H;
  const float* row = xs + (size_t)n * H * HID + h * HID;
  float ss = 0.f, sd = 0.f;
  for (int c = 0; c < HID; ++c) {
    float v = row[c];
    ss += v * a_src[h * HID + c];
    sd += v * a_dst[h * HID + c];
  }
  s_src[t] = ss;
  s_dst[t] = sd;
}

// s_e[e,h] = ea_full[e,:] @ V[:,h]
__global__ void k_se(const float* __restrict__ ea, const float* __restrict__ loop,
                     const float* __restrict__ V, float* s_e, int H) {
  int e = blockIdx.x * blockDim.x + threadIdx.x;
  if (e >= E_TOT) return;
  float a[EDGE_DIM];
#pragma unroll
  for (int d = 0; d < EDGE_DIM; ++d) a[d] = eattr(ea, loop, e, d);
  for (int h = 0; h < H; ++h) {
    float s = 0.f;
#pragma unroll
    for (int d = 0; d < EDGE_DIM; ++d) s += a[d] * V[d * H + h];
    s_e[(size_t)e * H + h] = s;
  }
}

__device__ __forceinline__ float edge_score(const float* s_src, const float* s_dst,
                                            const float* s_e, int s, int d, int e, int h, int H) {
  return lrelu(s_src[s * H + h] + s_dst[d * H + h] + s_e[(size_t)e * H + h]);
}

// pass 1: segment max per (dst, head)
__global__ void k_score_max(const float* __restrict__ s_src, const float* __restrict__ s_dst,
                            const float* __restrict__ s_e, const int* __restrict__ ei,
                            unsigned* mkey, int H) {
  long long t = (long long)blockIdx.x * blockDim.x + threadIdx.x;
  if (t >= (long long)E_TOT * H) return;
  int e = (int)(t / H), h = (int)(t - (long long)e * H);
  int s = esrc(ei, e), d = edst(ei, e);
  atomicMax(&mkey[d * H + h], f2ord(edge_score(s_src, s_dst, s_e, s, d, e, h, H)));
}

// pass 2: segment sum of exp(score - max)
__global__ void k_exp_sum(const float* __restrict__ s_src, const float* __restrict__ s_dst,
                          const float* __restrict__ s_e, const int* __restrict__ ei,
                          const unsigned* __restrict__ mkey, float* ssum, int H) {
  long long t = (long long)blockIdx.x * blockDim.x + threadIdx.x;
  if (t >= (long long)E_TOT * H) return;
  int e = (int)(t / H), h = (int)(t - (long long)e * H);
  int s = esrc(ei, e), d = edst(ei, e);
  float sc = edge_score(s_src, s_dst, s_e, s, d, e, h, H);
  atomicAdd(&ssum[d * H + h], __expf(sc - ord2f(mkey[d * H + h])));
}

// pass 3: out[dst,h,c] += xs[src,h,c] * alpha(e,h). Threads: (edge, channel) -> coalesced
// 1KB gathers from xs / 1KB scatters to agg; both arrays fit in the 192MB L2.
__global__ void k_aggregate(const float* __restrict__ xs, const float* __restrict__ s_src,
                            const float* __restrict__ s_dst, const float* __restrict__ s_e,
                            const unsigned* __restrict__ mkey, const float* __restrict__ ssum,
                            const int* __restrict__ ei, float* agg, int H) {
  const int HC = H * HID;
  long long t = (long long)blockIdx.x * blockDim.x + threadIdx.x;
  if (t >= (long long)E_TOT * HC) return;
  int e = (int)(t / HC), c = (int)(t - (long long)e * HC);
  int h = c >> 6;  // HID == 64
  int s = esrc(ei, e), d = edst(ei, e);
  float sc = edge_score(s_src, s_dst, s_e, s, d, e, h, H);
  float alpha = __expf(sc - ord2f(mkey[d * H + h])) / (ssum[d * H + h] + 1e-16f);
  atomicAdd(&agg[(size_t)d * HC + c], xs[(size_t)s * HC + c] * alpha);
}

__global__ void k_bias_relu(float* h, const float* __restrict__ b, int C) {
  long long t = (long long)blockIdx.x * blockDim.x + threadIdx.x;
  if (t >= (long long)N_NODES * C) return;
  int c = (int)(t % C);
  h[t] = fmaxf(h[t] + b[c], 0.f);
}

// head-mean + b2, then per-graph atomic max (readout)
__global__ void k_readout_max(const float* __restrict__ agg2, const float* __restrict__ b2,
                              const int* __restrict__ batch, unsigned* gkey) {
  int t = blockIdx.x * blockDim.x + threadIdx.x;
  if (t >= N_NODES * HID) return;
  int n = t / HID, c = t - n * HID;
  float v = 0.5f * (agg2[(size_t)n * (H2 * HID) + c] + agg2[(size_t)n * (H2 * HID) + HID + c]) + b2[c];
  atomicMax(&gkey[batch[n] * HID + c], f2ord(v));
}

__global__ void k_final(const unsigned* __restrict__ gkey, const float* __restrict__ Wr,
                        const float* __restrict__ br, float* out) {
  int g = blockIdx.x * blockDim.x + threadIdx.x;
  if (g >= N_GRAPHS) return;
  float s = 0.f;
  for (int c = 0; c < HID; ++c) s += ord2f(gkey[g * HID + c]) * Wr[c];
  out[g] = s + br[0];
}

// ---------------- host launch ----------------
static inline int nblk(long long n, int b) { return (int)((n + b - 1) / b); }

extern "C" void kernel_launch(void* const* d_in, const int* in_sizes, int n_in,
                              void* d_out, int out_size, void* d_ws, size_t ws_size,
                              hipStream_t stream) {
  const float* x      = (const float*)d_in[0];
  const float* ea     = (const float*)d_in[1];
  const int*   ei     = (const int*)d_in[2];
  const int*   batch  = (const int*)d_in[3];
  const float* W1     = (const float*)d_in[4];
  const float* a_src1 = (const float*)d_in[5];
  const float* a_dst1 = (const float*)d_in[6];
  const float* We1    = (const float*)d_in[7];
  const float* a_e1   = (const float*)d_in[8];
  const float* b1     = (const float*)d_in[9];
  const float* W2     = (const float*)d_in[10];
  const float* a_src2 = (const float*)d_in[11];
  const float* a_dst2 = (const float*)d_in[12];
  const float* We2    = (const float*)d_in[13];
  const float* a_e2   = (const float*)d_in[14];
  const float* b2     = (const float*)d_in[15];
  const float* Wr     = (const float*)d_in[16];
  const float* br     = (const float*)d_in[17];
  float* out = (float*)d_out;

  // workspace layout (4-byte words). Zeroed region first so one kernel clears it.
  char* w = (char*)d_ws;
  size_t off = 0;
  auto alloc = [&](size_t elems) -> void* { void* p = w + off; off += elems * 4; return p; };
  float*    cnt   = (float*)   alloc(N_NODES);
  float*    loop  = (float*)   alloc((size_t)N_NODES * EDGE_DIM);
  unsigned* mkey1 = (unsigned*)alloc((size_t)N_NODES * H1);
  float*    ssum1 = (float*)   alloc((size_t)N_NODES * H1);
  float*    agg1  = (float*)   alloc((size_t)N_NODES * H1 * HID);
  unsigned* mkey2 = (unsigned*)alloc((size_t)N_NODES * H2);
  float*    ssum2 = (float*)   alloc((size_t)N_NODES * H2);
  float*    agg2  = (float*)   alloc((size_t)N_NODES * H2 * HID);
  unsigned* gkey  = (unsigned*)alloc((size_t)N_GRAPHS * HID);
  long long zero_words = (long long)(off / 4);
  float* xs1   = (float*)alloc((size_t)N_NODES * H1 * HID);
  float* ssrc1 = (float*)alloc((size_t)N_NODES * H1);
  float* sdst1 = (float*)alloc((size_t)N_NODES * H1);
  float* V1    = (float*)alloc(EDGE_DIM * H1);
  float* se1   = (float*)alloc((size_t)E_TOT * H1);
  float* xs2   = (float*)alloc((size_t)N_NODES * H2 * HID);
  float* ssrc2 = (float*)alloc((size_t)N_NODES * H2);
  float* sdst2 = (float*)alloc((size_t)N_NODES * H2);
  float* V2    = (float*)alloc(EDGE_DIM * H2);
  float* se2   = (float*)alloc((size_t)E_TOT * H2);

  const int B = 256;

  // 0) clear accumulators
  k_zero<<<4096, B, 0, stream>>>((unsigned*)d_ws, zero_words);

  // 1) self-loop attrs (per-dst mean)
  k_loop_accum<<<nblk((long long)N_EDGES * EDGE_DIM, B), B, 0, stream>>>(ea, ei, cnt, loop);
  k_loop_div<<<nblk((long long)N_NODES * EDGE_DIM, B), B, 0, stream>>>(loop, cnt);

  // ---------------- layer 1 ----------------
  {
    long long waves = (long long)(N_NODES / 16) * ((H1 * HID) / 16);
    k_wmma_gemm_f32<<<nblk(waves * 32, B), B, 0, stream>>>(x, W1, xs1, N_NODES, H1 * HID, F_IN);
  }
  k_fold<<<1, EDGE_DIM * H1, 0, stream>>>(We1, a_e1, V1, H1);
  k_sdots<<<nblk((long long)N_NODES * H1, B), B, 0, stream>>>(xs1, a_src1, a_dst1, ssrc1, sdst1, H1);
  k_se<<<nblk(E_TOT, B), B, 0, stream>>>(ea, loop, V1, se1, H1);
  k_score_max<<<nblk((long long)E_TOT * H1, B), B, 0, stream>>>(ssrc1, sdst1, se1, ei, mkey1, H1);
  k_exp_sum<<<nblk((long long)E_TOT * H1, B), B, 0, stream>>>(ssrc1, sdst1, se1, ei, mkey1, ssum1, H1);
  k_aggregate<<<nblk((long long)E_TOT * H1 * HID, B), B, 0, stream>>>(xs1, ssrc1, sdst1, se1,
                                                                      mkey1, ssum1, ei, agg1, H1);
  k_bias_relu<<<nblk((long long)N_NODES * H1 * HID, B), B, 0, stream>>>(agg1, b1, H1 * HID);

  // ---------------- layer 2 ----------------
  {
    long long waves = (long long)(N_NODES / 16) * ((H2 * HID) / 16);
    k_wmma_gemm_f32<<<nblk(waves * 32, B), B, 0, stream>>>(agg1, W2, xs2, N_NODES, H2 * HID, H1 * HID);
  }
  k_fold<<<1, EDGE_DIM * H2, 0, stream>>>(We2, a_e2, V2, H2);
  k_sdots<<<nblk((long long)N_NODES * H2, B), B, 0, stream>>>(xs2, a_src2, a_dst2, ssrc2, sdst2, H2);
  k_se<<<nblk(E_TOT, B), B, 0, stream>>>(ea, loop, V2, se2, H2);
  k_score_max<<<nblk((long long)E_TOT * H2, B), B, 0, stream>>>(ssrc2, sdst2, se2, ei, mkey2, H2);
  k_exp_sum<<<nblk((long long)E_TOT * H2, B), B, 0, stream>>>(ssrc2, sdst2, se2, ei, mkey2, ssum2, H2);
  k_aggregate<<<nblk((long long)E_TOT * H2 * HID, B), B, 0, stream>>>(xs2, ssrc2, sdst2, se2,
                                                                      mkey2, ssum2, ei, agg2, H2);

  // ---------------- readout ----------------
  k_readout_max<<<nblk((long long)N_NODES * HID, B), B, 0, stream>>>(agg2, b2, batch, gkey);
  k_final<<<1, N_GRAPHS, 0, stream>>>(gkey, Wr, br, out);
}